// GCN_48430051229905
// MI455X (gfx1250) — compile-verified
//
#include <hip/hip_runtime.h>
#include <math.h>

// GCN on MI455X (gfx1250): bandwidth-bound (~230MB traffic, ~10us floor @23.3TB/s).
// GEMMs use V_WMMA_F32_16X16X4_F32 (fp32 WMMA, wave32). Scatter-add stays in the
// 192MB L2 (all intermediates <= 16MB).

#define F_IN   256
#define F_HID  16
#define F_OUT  40

typedef float v2f __attribute__((ext_vector_type(2)));
typedef float v8f __attribute__((ext_vector_type(8)));

__device__ __forceinline__ v8f wmma_f32_16x16x4(v2f a, v2f b, v8f c) {
#if defined(__AMDGCN__) && __has_builtin(__builtin_amdgcn_wmma_f32_16x16x4_f32)
  // 8 args: (neg_a, A, neg_b, B, c_mod, C, reuse_a, reuse_b)
  return __builtin_amdgcn_wmma_f32_16x16x4_f32(false, a, false, b, (short)0, c,
                                               false, false);
#else
  return c; // host-compile pass only
#endif
}

// ---------------- degree / normalization ----------------

__global__ void k_init_deg(float* __restrict__ deg, int N) {
  int i = blockIdx.x * 256 + threadIdx.x;
  if (i < N) deg[i] = 1.0f;  // self-loop
}

__global__ void k_count_deg(const int* __restrict__ dst, float* __restrict__ deg,
                            int E) {
  int e = blockIdx.x * 256 + threadIdx.x;
  if (e < E) atomicAdd(&deg[dst[e]], 1.0f);
}

__global__ void k_dinv(float* __restrict__ deg, int N) {
  int i = blockIdx.x * 256 + threadIdx.x;
  if (i < N) deg[i] = rsqrtf(deg[i]);  // in-place: deg -> d^-1/2
}

// ---------------- layer 1: t1 = x @ W1 (WMMA fp32, K=256) ----------------
// One wave per 16-row tile. A: x[16 x 4] per step, lane = h*16+r holds
// x[m0+r][4k + 2h + j] in vgpr j. B: W1[4 x 16], vgpr j = W1[4k + 2h + j][r].
// C/D: vgpr i = row (i + 8h), col r. Requires N % 16 == 0 (100000 = 6250*16).

__global__ void k_gemm1(const float* __restrict__ x, const float* __restrict__ W1,
                        float* __restrict__ t1, int nTiles) {
  const int wid  = (int)((blockIdx.x * blockDim.x + threadIdx.x) >> 5);
  const int lane = (int)(threadIdx.x & 31);
  if (wid >= nTiles) return;               // wave-uniform: EXEC all-1 for WMMA
  const int m0 = wid << 4;
  const int h  = lane >> 4;
  const int r  = lane & 15;
  const float* pa = x  + (size_t)(m0 + r) * F_IN + (h << 1);
  const float* pb = W1 + (size_t)(h << 1) * F_HID + r;
  v8f c = {};
#pragma unroll 8
  for (int k = 0; k < F_IN / 4; ++k) {
    v2f a, b;
    a.x = pa[0];
    a.y = pa[1];
    b.x = pb[0];
    b.y = pb[F_HID];
    c = wmma_f32_16x16x4(a, b, c);
    pa += 4;
    pb += 4 * F_HID;
  }
  float* po = t1 + (size_t)(m0 + (h << 3)) * F_HID + r;
#pragma unroll
  for (int i = 0; i < 8; ++i) po[i * F_HID] = c[i];
}

// h_init[n][f] = t1[n][f]*dinv[n]^2 + b1[f]   (self-loop term + bias)
__global__ void k_self1(const float* __restrict__ t1, const float* __restrict__ dinv,
                        const float* __restrict__ b1, float* __restrict__ hb, int N) {
  int i = blockIdx.x * 256 + threadIdx.x;
  if (i < N * F_HID) {
    int n = i >> 4, f = i & 15;
    float di = dinv[n];
    hb[i] = t1[i] * di * di + b1[f];
  }
}

// 16 lanes per edge; atomics hit L2 (hb = 6.4MB << 192MB L2)
__global__ void k_scatter1(const int* __restrict__ src, const int* __restrict__ dst,
                           const float* __restrict__ dinv, const float* __restrict__ t1,
                           float* __restrict__ hb, int E) {
  int g = blockIdx.x * 256 + threadIdx.x;
  int e = g >> 4, f = g & 15;
  if (e < E) {
    int s = src[e], d = dst[e];
    float w = dinv[s] * dinv[d];
    atomicAdd(&hb[(size_t)d * F_HID + f], t1[(size_t)s * F_HID + f] * w);
  }
}

__global__ void k_relu(float* __restrict__ hb, int n) {
  int i = blockIdx.x * 256 + threadIdx.x;
  if (i < n) hb[i] = fmaxf(hb[i], 0.0f);
}

// ---------------- layer 2: t2 = h @ W2 (WMMA fp32, K=16, N padded 40->48) ----
__global__ void k_gemm2(const float* __restrict__ hb, const float* __restrict__ W2,
                        float* __restrict__ t2, int nTiles) {
  const int wid  = (int)((blockIdx.x * blockDim.x + threadIdx.x) >> 5);
  const int lane = (int)(threadIdx.x & 31);
  if (wid >= nTiles) return;
  const int m0 = wid << 4;
  const int h  = lane >> 4;
  const int r  = lane & 15;
  const float* pa = hb + (size_t)(m0 + r) * F_HID + (h << 1);
  v2f a[4];
#pragma unroll
  for (int k = 0; k < 4; ++k) { a[k].x = pa[4 * k]; a[k].y = pa[4 * k + 1]; }
#pragma unroll
  for (int t = 0; t < 3; ++t) {
    const int col   = t * 16 + r;
    const int colc  = col < F_OUT ? col : (F_OUT - 1);  // clamp: no branch around WMMA
    const float msk = col < F_OUT ? 1.0f : 0.0f;
    v8f c = {};
#pragma unroll
    for (int k = 0; k < 4; ++k) {
      const int row0 = 4 * k + (h << 1);
      v2f b;
      b.x = W2[row0 * F_OUT + colc] * msk;
      b.y = W2[(row0 + 1) * F_OUT + colc] * msk;
      c = wmma_f32_16x16x4(a[k], b, c);
    }
    if (col < F_OUT) {
      float* po = t2 + (size_t)(m0 + (h << 3)) * F_OUT + col;
#pragma unroll
      for (int i = 0; i < 8; ++i) po[i * F_OUT] = c[i];
    }
  }
}

__global__ void k_self2(const float* __restrict__ t2, const float* __restrict__ dinv,
                        const float* __restrict__ b2, float* __restrict__ out, int N) {
  int i = blockIdx.x * 256 + threadIdx.x;
  if (i < N * F_OUT) {
    int n = i / F_OUT, f = i - n * F_OUT;
    float di = dinv[n];
    out[i] = t2[i] * di * di + b2[f];
  }
}

// 8 lanes per edge, 5 features each (40 total)
__global__ void k_scatter2(const int* __restrict__ src, const int* __restrict__ dst,
                           const float* __restrict__ dinv, const float* __restrict__ t2,
                           float* __restrict__ out, int E) {
  int g = blockIdx.x * 256 + threadIdx.x;
  int e = g >> 3, q = g & 7;
  if (e < E) {
    int s = src[e], d = dst[e];
    float w = dinv[s] * dinv[d];
    const float* ps = t2 + (size_t)s * F_OUT + q * 5;
    float* pd = out + (size_t)d * F_OUT + q * 5;
#pragma unroll
    for (int i = 0; i < 5; ++i) atomicAdd(&pd[i], ps[i] * w);
  }
}

// ---------------- log_softmax over 40 classes, one wave32 per row ----------
__global__ void k_logsoftmax(float* __restrict__ out, int N) {
  const int wid  = (int)((blockIdx.x * blockDim.x + threadIdx.x) >> 5);
  const int lane = (int)(threadIdx.x & 31);
  if (wid >= N) return;
  float* row = out + (size_t)wid * F_OUT;
  float a = row[lane];
  float b = (lane < F_OUT - 32) ? row[32 + lane] : -INFINITY;
  float m = fmaxf(a, b);
#pragma unroll
  for (int off = 16; off > 0; off >>= 1) m = fmaxf(m, __shfl_xor(m, off, 32));
  float s = __expf(a - m) + ((lane < F_OUT - 32) ? __expf(b - m) : 0.0f);
#pragma unroll
  for (int off = 16; off > 0; off >>= 1) s += __shfl_xor(s, off, 32);
  const float ls = m + __logf(s);
  row[lane] = a - ls;
  if (lane < F_OUT - 32) row[32 + lane] = b - ls;
}

// ---------------- launcher ----------------

extern "C" void kernel_launch(void* const* d_in, const int* in_sizes, int n_in,
                              void* d_out, int out_size, void* d_ws, size_t ws_size,
                              hipStream_t stream) {
  const float* x  = (const float*)d_in[0];
  const int*   ei = (const int*)d_in[1];   // [2, E] row-major: src row then dst row
  const float* W1 = (const float*)d_in[2];
  const float* b1 = (const float*)d_in[3];
  const float* W2 = (const float*)d_in[4];
  const float* b2 = (const float*)d_in[5];
  float* out = (float*)d_out;

  const int N = in_sizes[0] / F_IN;  // 100000
  const int E = in_sizes[1] / 2;     // 3200000
  const int* srcI = ei;
  const int* dstI = ei + E;

  // workspace layout (floats): dinv[N] | t1[N*16] | h[N*16] | t2[N*40]  (~29MB)
  float* ws   = (float*)d_ws;
  float* dinv = ws;
  float* t1   = dinv + N;
  float* hb   = t1 + (size_t)N * F_HID;
  float* t2   = hb + (size_t)N * F_HID;

  const int nTiles = N / 16;  // N divisible by 16
  dim3 B(256);
  auto grid = [](long long n) { return dim3((unsigned)((n + 255) / 256)); };

  k_init_deg  <<<grid(N),                 B, 0, stream>>>(dinv, N);
  k_count_deg <<<grid(E),                 B, 0, stream>>>(dstI, dinv, E);
  k_dinv      <<<grid(N),                 B, 0, stream>>>(dinv, N);

  k_gemm1     <<<grid((long long)nTiles * 32), B, 0, stream>>>(x, W1, t1, nTiles);
  k_self1     <<<grid((long long)N * F_HID),   B, 0, stream>>>(t1, dinv, b1, hb, N);
  k_scatter1  <<<grid((long long)E * 16),      B, 0, stream>>>(srcI, dstI, dinv, t1, hb, E);
  k_relu      <<<grid((long long)N * F_HID),   B, 0, stream>>>(hb, N * F_HID);

  k_gemm2     <<<grid((long long)nTiles * 32), B, 0, stream>>>(hb, W2, t2, nTiles);
  k_self2     <<<grid((long long)N * F_OUT),   B, 0, stream>>>(t2, dinv, b2, out, N);
  k_scatter2  <<<grid((long long)E * 8),       B, 0, stream>>>(srcI, dstI, dinv, t2, out, E);

  k_logsoftmax<<<grid((long long)N * 32),      B, 0, stream>>>(out, N);
}